// GraphTransformer_25039659336347
// MI455X (gfx1250) — compile-verified
//
#include <hip/hip_runtime.h>
#include <math.h>

// ---- problem constants (match reference) ----
#define BDIM    4
#define NNODES  2048
#define FDIM    64
#define HDIM    128
#define NHEADS  8
#define NLAYERS 3
#define NEDGES  32768
#define FFN     512

typedef __attribute__((ext_vector_type(2))) float v2f;
typedef __attribute__((ext_vector_type(8))) float v8f;

// CDNA5 fp32 WMMA: D = A(16x4) * B(4x16) + C(16x16), wave32.
__device__ __forceinline__ v8f wmma4(v2f a, v2f b, v8f c) {
  return __builtin_amdgcn_wmma_f32_16x16x4_f32(false, a, false, b, (short)0, c,
                                               false, false);
}

__device__ __forceinline__ float red_sum16(float v) {
#pragma unroll
  for (int m = 1; m < 16; m <<= 1) v += __shfl_xor(v, m, 32);
  return v;
}
__device__ __forceinline__ float red_max16(float v) {
#pragma unroll
  for (int m = 1; m < 16; m <<= 1) v = fmaxf(v, __shfl_xor(v, m, 32));
  return v;
}
__device__ __forceinline__ float red_sum32(float v) {
#pragma unroll
  for (int m = 1; m < 32; m <<= 1) v += __shfl_xor(v, m, 32);
  return v;
}

// -------- generic fp32 WMMA GEMM: C[M,N] = act(A[M,K] @ B[K,N]) --------------
// 4 waves per block; each wave owns a 16x32 output tile (two f32 accumulators
// sharing one A fragment -> 2 WMMA per A load, halves A traffic vs 16x16).
// A-frag layout (ISA 16x4 f32): lanes 0-15 hold M=lane K={0,1}; lanes 16-31 K={2,3}.
// C/D layout: VGPR r -> (M = r + 8*(lane>=16), N = lane&15).
template <int ACT>
__global__ void wmma_gemm_kernel(const float* __restrict__ A,
                                 const float* __restrict__ B,
                                 float* __restrict__ C, int M, int N, int K) {
  const int ntn  = N >> 5;  // 32-wide column tiles
  const int tile = blockIdx.x * 4 + (threadIdx.x >> 5);
  const int row0 = (tile / ntn) << 4;
  const int col0 = (tile - (tile / ntn) * ntn) << 5;
  const int lane = threadIdx.x & 31;
  const int m    = lane & 15;
  const int half = lane >> 4;
  const int koff = half << 1;
  v8f acc0 = {}, acc1 = {};
  const float* arow = A + (size_t)(row0 + m) * K;
  for (int k0 = 0; k0 < K; k0 += 4) {
    // pull the next A K-block toward the WGP while the WMMAs drain
    __builtin_prefetch(arow + k0 + 32, 0, 1);
    v2f a, b0, b1;
    a.x = arow[k0 + koff];
    a.y = arow[k0 + koff + 1];
    const float* br0 = B + (size_t)(k0 + koff) * N + col0 + m;
    const float* br1 = B + (size_t)(k0 + koff + 1) * N + col0 + m;
    b0.x = br0[0];
    b0.y = br1[0];
    b1.x = br0[16];
    b1.y = br1[16];
    acc0 = wmma4(a, b0, acc0);
    acc1 = wmma4(a, b1, acc1);
  }
  const int mb = half << 3;
#pragma unroll
  for (int r = 0; r < 8; ++r) {
    float v0 = acc0[r];
    float v1 = acc1[r];
    if (ACT == 1) {
      v0 = fmaxf(v0, 0.0f);
      v1 = fmaxf(v1, 0.0f);
    }
    float* crow = C + (size_t)(row0 + mb + r) * N + col0 + m;
    crow[0]  = v0;
    crow[16] = v1;
  }
}

// ---------------- positional encoding add (computed on the fly) --------------
__global__ void add_pe_kernel(float* __restrict__ x, const float* __restrict__ pos) {
  const int i = blockIdx.x * blockDim.x + threadIdx.x;
  if (i >= BDIM * NNODES * HDIM) return;
  const int hcol = i & (HDIM - 1);
  const int bn   = i >> 7;
  int p = (int)(pos[bn] * 999.0f);
  p = min(max(p, 0), 999);
  const int   he  = hcol & ~1;
  const float div = __expf((float)he * (-9.2103403719761836f / (float)HDIM));
  const float ang = (float)p * div;
  x[i] += (hcol & 1) ? __cosf(ang) : __sinf(ang);
}

// ---------------- adjacency bitmask ------------------------------------------
__global__ void adj_zero_kernel(unsigned* adjw) {
  const int i = blockIdx.x * blockDim.x + threadIdx.x;
  if (i < NNODES * 64) adjw[i] = 0u;
}
__global__ void adj_scatter_kernel(const int* __restrict__ ei, unsigned* adjw) {
  const int e = blockIdx.x * blockDim.x + threadIdx.x;
  if (e >= NEDGES) return;
  const int s = ei[e];
  const int t = ei[NEDGES + e];
  atomicOr(&adjw[s * 64 + (t >> 5)], 1u << (t & 31));
}

// ---------------- masked attention, one (b, head, 16-row tile) per block -----
// 8 waves; 16x2048 fp32 score stripe lives entirely in LDS (131 KB of the
// WGP's 320 KB). Phase1: S = QK^T via WMMA, scale+mask in C-frag epilogue.
// Phase2: row softmax stats (16-lane shuffle reductions). Phase3: O = P@V via
// WMMA, cross-wave reduction through LDS, normalize by row sum (0 if no edges).
__global__ void attention_kernel(const float* __restrict__ q,
                                 const float* __restrict__ k,
                                 const float* __restrict__ v,
                                 const unsigned* __restrict__ adjw,
                                 float* __restrict__ out) {
  extern __shared__ float smem[];
  float* sc   = smem;               // 16 * 2048
  float* rmax = smem + 16 * NNODES; // 16
  float* rsum = rmax + 16;          // 16

  const int idx  = blockIdx.x;
  const int rt   = idx & 127;
  const int h    = (idx >> 7) & 7;
  const int b    = idx >> 10;
  const int row0 = rt << 4;
  const int h16  = h << 4;
  const int tid  = threadIdx.x;
  const int w    = tid >> 5;
  const int lane = tid & 31;
  const int m    = lane & 15;
  const int half = lane >> 4;
  const int koff = half << 1;
  const int mb   = half << 3;

  const float* qb  = q + (size_t)b * NNODES * HDIM;
  const float* kbp = k + (size_t)b * NNODES * HDIM;
  const float* vbp = v + (size_t)b * NNODES * HDIM;

  // hoist Q fragments (head dim 16 -> 4 k-steps of 4)
  v2f aq[4];
#pragma unroll
  for (int ks = 0; ks < 4; ++ks) {
    const int c0 = (ks << 2) + koff;
    aq[ks].x = qb[(size_t)(row0 + m) * HDIM + h16 + c0];
    aq[ks].y = qb[(size_t)(row0 + m) * HDIM + h16 + c0 + 1];
  }

  // phase 1: scores -> LDS (scaled, adjacency-masked)
  for (int jt = w; jt < NNODES / 16; jt += 8) {
    const int col0 = jt << 4;
    v8f acc = {};
#pragma unroll
    for (int ks = 0; ks < 4; ++ks) {
      const int kabs = (ks << 2) + koff;
      v2f bb;
      bb.x = kbp[(size_t)(col0 + m) * HDIM + h16 + kabs];
      bb.y = kbp[(size_t)(col0 + m) * HDIM + h16 + kabs + 1];
      acc = wmma4(aq[ks], bb, acc);
    }
#pragma unroll
    for (int r = 0; r < 8; ++r) {
      const int mm       = mb + r;
      const int colg     = col0 + m;
      const unsigned wd  = adjw[(size_t)(row0 + mm) * 64 + (colg >> 5)];
      const float sv = ((wd >> (colg & 31)) & 1u) ? acc[r] * 0.25f : -INFINITY;
      sc[mm * NNODES + colg] = sv;
    }
  }
  __syncthreads();

  // phase 2: softmax (store exp back, keep row sums for late normalization)
  {
    const int row = tid >> 4;
    const int cc  = tid & 15;
    float mx = -INFINITY;
    for (int col = cc; col < NNODES; col += 16)
      mx = fmaxf(mx, sc[row * NNODES + col]);
    mx = red_max16(mx);
    if (cc == 0) rmax[row] = mx;
    __syncthreads();
    const float M = rmax[row];
    float sm = 0.0f;
    for (int col = cc; col < NNODES; col += 16) {
      const float s = sc[row * NNODES + col];
      const float e = (M == -INFINITY) ? 0.0f : __expf(s - M);
      sc[row * NNODES + col] = e;
      sm += e;
    }
    sm = red_sum16(sm);
    if (cc == 0) rsum[row] = sm;
  }
  __syncthreads();

  // phase 3: O = P @ V, K split across the 8 waves
  v8f co = {};
  for (int jt = w; jt < NNODES / 16; jt += 8) {
    const int c0 = jt << 4;
#pragma unroll
    for (int ks = 0; ks < 4; ++ks) {
      const int ca = (ks << 2) + koff;
      v2f ap, bv;
      ap.x = sc[m * NNODES + c0 + ca];
      ap.y = sc[m * NNODES + c0 + ca + 1];
      bv.x = vbp[(size_t)(c0 + ca) * HDIM + h16 + m];
      bv.y = vbp[(size_t)(c0 + ca + 1) * HDIM + h16 + m];
      co = wmma4(ap, bv, co);
    }
  }
  __syncthreads();  // all waves done reading P -> safe to reuse LDS
#pragma unroll
  for (int r = 0; r < 8; ++r) sc[w * 256 + (mb + r) * 16 + m] = co[r];
  __syncthreads();
  {
    const int mm = tid >> 4;
    const int nn = tid & 15;
    float acc = 0.0f;
#pragma unroll
    for (int ww = 0; ww < 8; ++ww) acc += sc[ww * 256 + tid];
    const float rs  = rsum[mm];
    const float inv = (rs > 0.0f) ? 1.0f / rs : 0.0f;  // no-edge rows -> 0
    out[(size_t)b * NNODES * HDIM + (size_t)(row0 + mm) * HDIM + h16 + nn] =
        acc * inv;
  }
}

// ---------------- LayerNorm(a + r) -------------------------------------------
__global__ void add_ln_kernel(const float* __restrict__ A,
                              const float* __restrict__ R,
                              const float* __restrict__ g,
                              const float* __restrict__ be,
                              float* __restrict__ out, int rows) {
  const int wid  = (blockIdx.x * blockDim.x + threadIdx.x) >> 5;
  const int lane = threadIdx.x & 31;
  if (wid >= rows) return;
  const float* a = A + (size_t)wid * HDIM;
  const float* r = R + (size_t)wid * HDIM;
  float vals[4];
  float s = 0.f, sq = 0.f;
#pragma unroll
  for (int i = 0; i < 4; ++i) {
    const int c  = lane + (i << 5);
    const float vv = a[c] + r[c];
    vals[i] = vv; s += vv; sq += vv * vv;
  }
  s = red_sum32(s); sq = red_sum32(sq);
  const float mean = s * (1.0f / HDIM);
  const float var  = sq * (1.0f / HDIM) - mean * mean;
  const float rstd = rsqrtf(var + 1e-5f);
  float* o = out + (size_t)wid * HDIM;
#pragma unroll
  for (int i = 0; i < 4; ++i) {
    const int c = lane + (i << 5);
    o[c] = (vals[i] - mean) * rstd * g[c] + be[c];
  }
}

// -------- fused pooling MLP: logits = w2 . tanh(A @ W1 + b1) + b2 ------------
// One wave per 16 rows; hidden activations stay in WMMA accumulators.
// gather=1: A row e = concat(x[b,src[e]], x[b,tgt[e]]) (K=256).
__global__ void mlp_logit_kernel(const float* __restrict__ X,
                                 const int* __restrict__ srcI,
                                 const int* __restrict__ tgtI,
                                 const float* __restrict__ W1,
                                 const float* __restrict__ b1,
                                 const float* __restrict__ w2,
                                 const float* __restrict__ b2,
                                 float* __restrict__ logits, int K, int Hn,
                                 int rowsPerB, int gather) {
  const int row0 = blockIdx.x << 4;
  const int lane = threadIdx.x;
  const int m    = lane & 15;
  const int half = lane >> 4;
  const int koff = half << 1;
  const int mb   = half << 3;
  const int myrow = row0 + m;
  const float* rA;
  const float* rB = nullptr;
  if (gather) {
    const int bb = myrow / rowsPerB;
    const int e  = myrow - bb * rowsPerB;
    rA = X + (size_t)(bb * NNODES + srcI[e]) * HDIM;
    rB = X + (size_t)(bb * NNODES + tgtI[e]) * HDIM;
  } else {
    rA = X + (size_t)myrow * K;
  }
  float part[8];
#pragma unroll
  for (int r = 0; r < 8; ++r) part[r] = 0.f;
  for (int nt = 0; nt < Hn; nt += 16) {
    v8f acc = {};
    for (int k0 = 0; k0 < K; k0 += 4) {
      const int c0 = k0 + koff;
      v2f a, bfr;
      if (gather) {
        a.x = (c0 < HDIM) ? rA[c0] : rB[c0 - HDIM];
        a.y = (c0 + 1 < HDIM) ? rA[c0 + 1] : rB[c0 + 1 - HDIM];
      } else {
        a.x = rA[c0];
        a.y = rA[c0 + 1];
      }
      bfr.x = W1[(size_t)c0 * Hn + nt + m];
      bfr.y = W1[(size_t)(c0 + 1) * Hn + nt + m];
      acc = wmma4(a, bfr, acc);
    }
    const int col   = nt + m;
    const float b1c = b1[col];
    const float w2c = w2[col];
#pragma unroll
    for (int r = 0; r < 8; ++r) part[r] += tanhf(acc[r] + b1c) * w2c;
  }
  const float b2v = b2[0];
#pragma unroll
  for (int r = 0; r < 8; ++r) {
    const float s = red_sum16(part[r]);
    if (m == 0) logits[row0 + mb + r] = s + b2v;
  }
}

// ---------------- softmax over one axis (stats + normalize) ------------------
__global__ void softmax_stats_kernel(const float* __restrict__ logit, int len,
                                     float* __restrict__ maxv,
                                     float* __restrict__ sumv) {
  const int b = blockIdx.x;
  const float* p = logit + (size_t)b * len;
  __shared__ float sh[8];
  __shared__ float shmax;
  const int tid = threadIdx.x;
  float mx = -INFINITY;
  for (int i = tid; i < len; i += 256) mx = fmaxf(mx, p[i]);
  for (int msk = 1; msk < 32; msk <<= 1) mx = fmaxf(mx, __shfl_xor(mx, msk, 32));
  if ((tid & 31) == 0) sh[tid >> 5] = mx;
  __syncthreads();
  if (tid == 0) {
    float vv = sh[0];
    for (int i2 = 1; i2 < 8; ++i2) vv = fmaxf(vv, sh[i2]);
    shmax = vv;
  }
  __syncthreads();
  const float M = shmax;
  float sm = 0.f;
  for (int i = tid; i < len; i += 256) sm += __expf(p[i] - M);
  sm = red_sum32(sm);
  if ((tid & 31) == 0) sh[tid >> 5] = sm;
  __syncthreads();
  if (tid == 0) {
    float vv = 0.f;
    for (int i2 = 0; i2 < 8; ++i2) vv += sh[i2];
    maxv[b] = M;
    sumv[b] = vv;
  }
}
__global__ void softmax_norm_kernel(float* __restrict__ logit, int total, int len,
                                    const float* __restrict__ maxv,
                                    const float* __restrict__ sumv) {
  const int i = blockIdx.x * blockDim.x + threadIdx.x;
  if (i >= total) return;
  const int b = i / len;
  logit[i] = __expf(logit[i] - maxv[b]) / sumv[b];
}

// ---------------- pooling reductions -----------------------------------------
__global__ void pool_nodes_kernel(const float* __restrict__ prob,
                                  const float* __restrict__ x,
                                  float* __restrict__ npool) {
  const int b    = blockIdx.x;
  const int hcol = threadIdx.x;  // 128
  float acc = 0.f;
  for (int n = 0; n < NNODES; ++n)
    acc += prob[b * NNODES + n] * x[(size_t)(b * NNODES + n) * HDIM + hcol];
  npool[b * HDIM + hcol] = acc;
}
__global__ void rowsum_kernel(const float* __restrict__ x, float* __restrict__ rs,
                              int rows) {
  const int wid  = (blockIdx.x * blockDim.x + threadIdx.x) >> 5;
  const int lane = threadIdx.x & 31;
  if (wid >= rows) return;
  const float* p = x + (size_t)wid * HDIM;
  float s = 0.f;
#pragma unroll
  for (int i = 0; i < 4; ++i) s += p[lane + (i << 5)];
  s = red_sum32(s);
  if (lane == 0) rs[wid] = s;
}
__global__ void sp_kernel(const float* __restrict__ prob,
                          const float* __restrict__ rs,
                          const int* __restrict__ ei, float* __restrict__ sp) {
  const int b   = blockIdx.x;
  const int tid = threadIdx.x;
  __shared__ float sh[8];
  float acc = 0.f;
  for (int e = tid; e < NEDGES; e += 256) {
    const float mean =
        (rs[b * NNODES + ei[e]] + rs[b * NNODES + ei[NEDGES + e]]) *
        (1.0f / 256.0f);
    acc += prob[b * NEDGES + e] * mean;
  }
  acc = red_sum32(acc);
  if ((tid & 31) == 0) sh[tid >> 5] = acc;
  __syncthreads();
  if (tid == 0) {
    float vv = 0.f;
    for (int i = 0; i < 8; ++i) vv += sh[i];
    sp[b] = vv;
  }
}

// ---------------- output assembly --------------------------------------------
__global__ void write_emb_kernel(const float* __restrict__ npool,
                                 const float* __restrict__ sp,
                                 float* __restrict__ out) {
  const int i = blockIdx.x * blockDim.x + threadIdx.x;
  if (i >= BDIM * 256) return;
  const int b = i >> 8;
  const int j = i & 255;
  out[i] = (j < HDIM) ? npool[b * HDIM + j] : sp[b];
}
__global__ void copy_nodes_kernel(const float* __restrict__ x,
                                  float* __restrict__ out) {
  const int i = blockIdx.x * blockDim.x + threadIdx.x;
  if (i < BDIM * NNODES * HDIM) out[BDIM * 256 + i] = x[i];
}

// =============================================================================
extern "C" void kernel_launch(void* const* d_in, const int* in_sizes, int n_in,
                              void* d_out, int out_size, void* d_ws,
                              size_t ws_size, hipStream_t stream) {
  (void)in_sizes; (void)n_in; (void)out_size; (void)ws_size;
  const float* node_features = (const float*)d_in[0];
  const int*   edge_index    = (const int*)d_in[1];
  /* d_in[2] edge_weights: unused by the reference */
  const float* layer_pos = (const float*)d_in[3];
  const float* w_in      = (const float*)d_in[4];
  const float* Wq        = (const float*)d_in[5];
  const float* Wk        = (const float*)d_in[6];
  const float* Wv        = (const float*)d_in[7];
  const float* Wo        = (const float*)d_in[8];
  const float* ln_a_g    = (const float*)d_in[9];
  const float* ln_a_b    = (const float*)d_in[10];
  const float* ln_o_g    = (const float*)d_in[11];
  const float* ln_o_b    = (const float*)d_in[12];
  const float* ffn_w1    = (const float*)d_in[13];
  const float* ffn_w2    = (const float*)d_in[14];
  const float* na_w1     = (const float*)d_in[15];
  const float* na_b1     = (const float*)d_in[16];
  const float* na_w2     = (const float*)d_in[17];
  const float* na_b2     = (const float*)d_in[18];
  const float* sa_w1     = (const float*)d_in[19];
  const float* sa_b1     = (const float*)d_in[20];
  const float* sa_w2     = (const float*)d_in[21];
  const float* sa_b2     = (const float*)d_in[22];
  float* out = (float*)d_out;

  const size_t NT = (size_t)BDIM * NNODES * HDIM;  // 1,048,576
  float*    xb     = (float*)d_ws;
  float*    qb     = xb + NT;
  float*    kb     = qb + NT;
  float*    vb     = kb + NT;
  float*    ao     = vb + NT;
  float*    fh     = ao + NT;                      // B*N*4H
  unsigned* adjw   = (unsigned*)(fh + 4 * NT);     // 2048*64 words
  float*    na_log = (float*)(adjw + NNODES * 64); // B*N
  float*    rsb    = na_log + BDIM * NNODES;       // B*N
  float*    sa_log = rsb + BDIM * NNODES;          // B*E
  float*    stats  = sa_log + BDIM * NEDGES;       // 16
  float*    npool  = stats + 16;                   // B*H
  float*    spb    = npool + BDIM * HDIM;          // B

  const int rows = BDIM * NNODES;  // 8192
  // GEMM tiling: 16x32 per wave, 4 waves per block
  const int g_h   = (rows / 16) * (HDIM / 32) / 4;  // N=128 outputs -> 512
  const int g_ffn = (rows / 16) * (FFN / 32) / 4;   // N=512 outputs -> 2048

  // input projection + positional encoding
  wmma_gemm_kernel<0><<<g_h, 128, 0, stream>>>(node_features, w_in, xb, rows,
                                               HDIM, FDIM);
  add_pe_kernel<<<(rows * HDIM) / 256, 256, 0, stream>>>(xb, layer_pos);

  // adjacency bitmask
  adj_zero_kernel<<<(NNODES * 64) / 256, 256, 0, stream>>>(adjw);
  adj_scatter_kernel<<<NEDGES / 256, 256, 0, stream>>>(edge_index, adjw);

  const size_t attn_lds = (size_t)(16 * NNODES + 32) * sizeof(float);  // 131 KB
  for (int l = 0; l < NLAYERS; ++l) {
    const float* Wql = Wq + (size_t)l * HDIM * HDIM;
    const float* Wkl = Wk + (size_t)l * HDIM * HDIM;
    const float* Wvl = Wv + (size_t)l * HDIM * HDIM;
    const float* Wol = Wo + (size_t)l * HDIM * HDIM;
    wmma_gemm_kernel<0><<<g_h, 128, 0, stream>>>(xb, Wql, qb, rows, HDIM, HDIM);
    wmma_gemm_kernel<0><<<g_h, 128, 0, stream>>>(xb, Wkl, kb, rows, HDIM, HDIM);
    wmma_gemm_kernel<0><<<g_h, 128, 0, stream>>>(xb, Wvl, vb, rows, HDIM, HDIM);
    attention_kernel<<<BDIM * NHEADS * (NNODES / 16), 256, attn_lds, stream>>>(
        qb, kb, vb, adjw, ao);
    wmma_gemm_kernel<0><<<g_h, 128, 0, stream>>>(ao, Wol, qb, rows, HDIM, HDIM);
    add_ln_kernel<<<1024, 256, 0, stream>>>(qb, xb, ln_a_g + l * HDIM,
                                            ln_a_b + l * HDIM, kb, rows);
    add_ln_kernel<<<1024, 256, 0, stream>>>(kb, xb, ln_o_g + l * HDIM,
                                            ln_o_b + l * HDIM, vb, rows);
    wmma_gemm_kernel<1><<<g_ffn, 128, 0, stream>>>(
        vb, ffn_w1 + (size_t)l * HDIM * FFN, fh, rows, FFN, HDIM);
    wmma_gemm_kernel<0><<<g_h, 128, 0, stream>>>(
        fh, ffn_w2 + (size_t)l * FFN * HDIM, qb, rows, HDIM, FFN);
    add_ln_kernel<<<1024, 256, 0, stream>>>(qb, vb, ln_o_g + l * HDIM,
                                            ln_o_b + l * HDIM, xb, rows);
  }

  // hierarchical pooling: node attention
  mlp_logit_kernel<<<rows / 16, 32, 0, stream>>>(
      xb, nullptr, nullptr, na_w1, na_b1, na_w2, na_b2, na_log, HDIM, HDIM / 2,
      NNODES, 0);
  softmax_stats_kernel<<<BDIM, 256, 0, stream>>>(na_log, NNODES, stats, stats + 4);
  softmax_norm_kernel<<<(BDIM * NNODES) / 256, 256, 0, stream>>>(
      na_log, BDIM * NNODES, NNODES, stats, stats + 4);
  pool_nodes_kernel<<<BDIM, HDIM, 0, stream>>>(na_log, xb, npool);

  // structure pooling over edges
  rowsum_kernel<<<1024, 256, 0, stream>>>(xb, rsb, rows);
  mlp_logit_kernel<<<(BDIM * NEDGES) / 16, 32, 0, stream>>>(
      xb, edge_index, edge_index + NEDGES, sa_w1, sa_b1, sa_w2, sa_b2, sa_log,
      2 * HDIM, HDIM, NEDGES, 1);
  softmax_stats_kernel<<<BDIM, 256, 0, stream>>>(sa_log, NEDGES, stats + 8,
                                                 stats + 12);
  softmax_norm_kernel<<<(BDIM * NEDGES) / 256, 256, 0, stream>>>(
      sa_log, BDIM * NEDGES, NEDGES, stats + 8, stats + 12);
  sp_kernel<<<BDIM, 256, 0, stream>>>(sa_log, rsb, edge_index, spb);

  // outputs: [B,2H] global embedding then [B,N,H] node embeddings
  write_emb_kernel<<<4, 256, 0, stream>>>(npool, spb, out);
  copy_nodes_kernel<<<(rows * HDIM) / 256, 256, 0, stream>>>(xb, out);
}